// ctc_layer_64553358459230
// MI455X (gfx1250) — compile-verified
//
#include <hip/hip_runtime.h>

// CTC greedy decode for x[64, 800, 1428] f32.
// Stage 1: wave32-per-row argmax over C (memory-bound: ~292 MB @ 23.3 TB/s ~ 12.5 us).
// Stage 2: wave32-per-batch stream compaction (merge repeats, drop blank, pad -1).

#define BB 64
#define TT 800
#define CC 1428
#define BLANK_ (CC - 1)
#define NROWS (BB * TT)

__global__ __launch_bounds__(256) void ctc_argmax_kernel(const float* __restrict__ x,
                                                         int* __restrict__ path) {
    const int wave = blockIdx.x * (blockDim.x >> 5) + (threadIdx.x >> 5);
    const int lane = threadIdx.x & 31;
    if (wave >= NROWS) return;

    const float* __restrict__ row = x + (size_t)wave * CC;

    // Pull the tail cacheline early (emits global_prefetch_b8 on gfx1250).
    __builtin_prefetch(row + 1408, 0, 0);

    float bestV = -__builtin_inff();
    int   bestI = 0;

    // 11 x 128 floats = 1408 elements, float4 per lane, fully unrolled so all
    // 11 independent global_load_b128 can be in flight at once.
#pragma unroll
    for (int i = 0; i < 11; ++i) {
        const int idx = i * 128 + lane * 4;
        const float4 v = *reinterpret_cast<const float4*>(row + idx);
        // strict '>' keeps the FIRST occurrence of the max within a lane
        if (v.x > bestV) { bestV = v.x; bestI = idx;     }
        if (v.y > bestV) { bestV = v.y; bestI = idx + 1; }
        if (v.z > bestV) { bestV = v.z; bestI = idx + 2; }
        if (v.w > bestV) { bestV = v.w; bestI = idx + 3; }
    }
    // tail: elements 1408..1427 (20 elements)
    if (lane < (CC - 1408)) {
        const int idx = 1408 + lane;
        const float v = row[idx];
        if (v > bestV) { bestV = v; bestI = idx; }
    }

    // wave32 butterfly reduction; on ties pick the smaller index (jnp.argmax rule)
#pragma unroll
    for (int off = 16; off >= 1; off >>= 1) {
        const float ov = __shfl_xor(bestV, off, 32);
        const int   oi = __shfl_xor(bestI, off, 32);
        if (ov > bestV || (ov == bestV && oi < bestI)) { bestV = ov; bestI = oi; }
    }

    if (lane == 0) path[wave] = bestI;
}

__global__ __launch_bounds__(256) void ctc_compact_kernel(const int* __restrict__ path,
                                                          float* __restrict__ out) {
    const int wave = blockIdx.x * (blockDim.x >> 5) + (threadIdx.x >> 5);
    const int lane = threadIdx.x & 31;
    if (wave >= BB) return;

    const int* __restrict__ p = path + wave * TT;
    float* __restrict__ o = out + (size_t)wave * TT;

    int base = 0;  // number of labels emitted so far for this batch row
#pragma unroll 1
    for (int c = 0; c < TT; c += 32) {
        const int t = c + lane;
        const int v = p[t];
        const int prev = (t == 0) ? -1 : p[t - 1];
        const bool keep = (v != prev) && (v != BLANK_);

        const unsigned m = (unsigned)__ballot(keep);          // wave32 mask
        const int offset = base + __popc(m & ((1u << lane) - 1u));
        if (keep) o[offset] = (float)v;
        base += __popc(m);
    }
    // pad remainder with -1
    for (int pos = base + lane; pos < TT; pos += 32) o[pos] = -1.0f;
}

extern "C" void kernel_launch(void* const* d_in, const int* in_sizes, int n_in,
                              void* d_out, int out_size, void* d_ws, size_t ws_size,
                              hipStream_t stream) {
    const float* x = (const float*)d_in[0];
    float* out = (float*)d_out;        // integer reference output -> float* per harness
    int* path = (int*)d_ws;            // 51200 * 4 B = 200 KB scratch

    // Stage 1: one wave per (b,t) row; 8 waves per 256-thread block.
    const int rowsPerBlock = 256 / 32;
    const int grid1 = (NROWS + rowsPerBlock - 1) / rowsPerBlock;   // 6400
    ctc_argmax_kernel<<<grid1, 256, 0, stream>>>(x, path);

    // Stage 2: one wave per batch row; 64 waves = 8 blocks of 256 threads.
    const int grid2 = (BB + rowsPerBlock - 1) / rowsPerBlock;      // 8
    ctc_compact_kernel<<<grid2, 256, 0, stream>>>(path, out);
}